// TkLastTradesAutoencoder_65197603553322
// MI455X (gfx1250) — compile-verified
//
#include <hip/hip_runtime.h>
#include <math.h>

// ---------------- problem constants ----------------
enum { B_ = 64, CIN_ = 8, L_ = 1024, D_ = 64, K_ = 256, E_ = 512 };
enum { CL_ = CIN_ * L_ };                 // 8192
enum { Y_SZ = B_ * CIN_ * L_ };           // 524288
enum { LV_SZ = B_ * B_ };                 // 4096
#define BETA_ 0.25f
#define EPS_LIP_ 0.01f

// ---------------- WMMA / TDM types ----------------
typedef float v2f __attribute__((ext_vector_type(2)));
typedef float v8f __attribute__((ext_vector_type(8)));
typedef unsigned v4u __attribute__((ext_vector_type(4)));
typedef int v4i __attribute__((ext_vector_type(4)));
typedef int v8i __attribute__((ext_vector_type(8)));

__device__ __forceinline__ v8f wmma_f32(v2f a, v2f b, v8f c) {
  // D = A(16x4 f32) * B(4x16 f32) + C(16x16 f32)
  return __builtin_amdgcn_wmma_f32_16x16x4_f32(false, a, false, b, (short)0, c,
                                               false, false);
}

// ---------------- TDM: DMA a strided 2D f32 tile (tile_x x tile_y) to LDS ----
// D# per CDNA5 ISA ch.8: group0 = {flags/count, lds_addr, global_addr, type},
// group1 = {data_size, tensor dims, tile dims, dim0 stride}. 2D tensor ->
// groups 2/3 zero. Issued by one wave; completion via s_wait_tensorcnt.
#if __has_builtin(__builtin_amdgcn_tensor_load_to_lds)
#define HAVE_TDM_ 1
__device__ __forceinline__ void tdm_load_tile_2d_f32(unsigned lds_off,
                                                     const void* gaddr,
                                                     int tile_x, int tile_y,
                                                     int stride_elems) {
  unsigned long long ga = (unsigned long long)gaddr;
  v4u g0;
  g0.x = 0x1u;                                          // count=1 (valid), user D#
  g0.y = lds_off;                                       // lds_addr (bytes)
  g0.z = (unsigned)ga;                                  // global_addr[31:0]
  g0.w = (unsigned)((ga >> 32) & 0x01FFFFFFu) | (2u << 30);  // addr[56:32] | type=2
  v8i g1;
  g1[0] = (2 << 16);                                    // data_size = 4 bytes
  g1[1] = (tile_x & 0xFFFF) << 16;                      // tensor_dim0[15:0]
  g1[2] = (tile_y & 0xFFFF) << 16;                      // dim0 hi=0 | tensor_dim1 lo
  g1[3] = (tile_x & 0xFFFF) << 16;                      // dim1 hi=0 | tile_dim0
  g1[4] = (tile_y & 0xFFFF);                            // tile_dim1 | tile_dim2=0
  g1[5] = stride_elems;                                 // tensor_dim0_stride lo32
  g1[6] = 0;
  g1[7] = 0;
  v4i zz = {0, 0, 0, 0};
#if __clang_major__ >= 23
  v8i z8 = {0, 0, 0, 0, 0, 0, 0, 0};
  __builtin_amdgcn_tensor_load_to_lds(g0, g1, zz, zz, z8, 0);
#else
  __builtin_amdgcn_tensor_load_to_lds(g0, g1, zz, zz, 0);
#endif
#if __has_builtin(__builtin_amdgcn_s_wait_tensorcnt)
  __builtin_amdgcn_s_wait_tensorcnt(0);
#endif
}
#else
#define HAVE_TDM_ 0
#endif

// ---------------- helpers ----------------
__device__ __forceinline__ float blockReduceSum(float v, float* sbuf) {
  int tid = threadIdx.x;
  sbuf[tid] = v;
  __syncthreads();
  for (int s = 128; s > 0; s >>= 1) {
    if (tid < s) sbuf[tid] += sbuf[tid + s];
    __syncthreads();
  }
  return sbuf[0];
}

// ---------------- conv1d, compile-time taps (cross-correlation, NCH/OIH) -----
template <int Cin, int Ks, int Stride, int Pad, int Relu>
__global__ void conv1d_t(const float* __restrict__ in, const float* __restrict__ w,
                         const float* __restrict__ bias, float* __restrict__ out,
                         int Lin, int Cout, int Lout) {
  int idx = blockIdx.x * blockDim.x + threadIdx.x;
  int total = B_ * Cout * Lout;
  if (idx >= total) return;
  int t = idx % Lout;
  int co = (idx / Lout) % Cout;
  int b = idx / (Lout * Cout);
  const float* ip = in + (long)b * Cin * Lin;
  const float* wp = w + (long)co * Cin * Ks;
  float acc = bias[co];
  for (int ci = 0; ci < Cin; ++ci) {
#pragma unroll
    for (int kk = 0; kk < Ks; ++kk) {
      int p = t * Stride + kk - Pad;
      if (p >= 0 && p < Lin) acc += ip[ci * Lin + p] * wp[ci * Ks + kk];
    }
  }
  if (Relu) acc = fmaxf(acc, 0.f);
  out[idx] = acc;
}

// ---------------- transposed conv1d via input-dilated correlation ------------
template <int Cin, int Ks, int Stride, int Pad, int Relu>
__global__ void convT1d_t(const float* __restrict__ in, const float* __restrict__ w,
                          const float* __restrict__ bias, float* __restrict__ out,
                          int Lin, int Cout, int Lout) {
  int idx = blockIdx.x * blockDim.x + threadIdx.x;
  int total = B_ * Cout * Lout;
  if (idx >= total) return;
  int t = idx % Lout;
  int co = (idx / Lout) % Cout;
  int b = idx / (Lout * Cout);
  const float* ip = in + (long)b * Cin * Lin;
  const float* wp = w + (long)co * Cin * Ks;
  const int lead = Ks - 1 - Pad;
  float acc = bias[co];
  for (int ci = 0; ci < Cin; ++ci) {
#pragma unroll
    for (int kk = 0; kk < Ks; ++kk) {
      int s0 = t + kk - lead;
      if (s0 >= 0 && (s0 % Stride) == 0) {
        int q = s0 / Stride;
        if (q < Lin) acc += ip[ci * Lin + q] * wp[ci * Ks + kk];
      }
    }
  }
  if (Relu) acc = fmaxf(acc, 0.f);
  out[idx] = acc;
}

// ---------------- row squared-norms ----------------
__global__ void rownorm_k(const float* __restrict__ m, float* __restrict__ out,
                          int rows, int cols) {
  int r = blockIdx.x * blockDim.x + threadIdx.x;
  if (r >= rows) return;
  const float* p = m + (long)r * cols;
  float s = 0.f;
  for (int c = 0; c < cols; ++c) s += p[c] * p[c];
  out[r] = s;
}

__global__ void rownorm_block_k(const float* __restrict__ m, float* __restrict__ out,
                                int cols) {
  __shared__ float sbuf[256];
  int r = blockIdx.x;
  float s = 0.f;
  for (int c = threadIdx.x; c < cols; c += 256) {
    float v = m[(long)r * cols + c];
    s += v * v;
  }
  float tot = blockReduceSum(s, sbuf);
  if (threadIdx.x == 0) out[r] = tot;
}

// ---------------- VQ nearest-code via WMMA f32 GEMM (zp @ cb^T) ----------------
// grid: 1024 blocks (16 rows each), 256 threads = 8 waves; each wave owns 4
// column tiles of 16 codes -> all 512 codes covered per block. The strided
// 16x64 A-tile is DMA'd into LDS by the Tensor Data Mover.
__global__ void vq_assign_k(const float* __restrict__ z, const float* __restrict__ cb,
                            const float* __restrict__ cnorm, int* __restrict__ codes) {
  __shared__ float sA[D_ * 16];   // sA[d*16 + row] = zp[row][d]
  __shared__ float sAnorm[16];
  __shared__ float sVal[8][16];
  __shared__ int sCol[8][16];

  int n0 = blockIdx.x * 16;          // flat row index into zp [B*K, D]
  int b = n0 / K_;
  int t0 = n0 % K_;
  const float* zb = z + (long)b * D_ * K_ + t0;  // zp[row][d] = zb[d*K_ + row]
  int tid = threadIdx.x;

#if HAVE_TDM_
  if (tid < 32) {
    // tile: x = 16 consecutive t (contiguous), y = 64 d rows, row stride = K_
    tdm_load_tile_2d_f32((unsigned)(unsigned long long)(void*)sA, (const void*)zb,
                         16, D_, K_);
  }
#else
  for (int i = tid; i < D_ * 16; i += 256) {
    int row = i & 15, d = i >> 4;
    sA[i] = zb[d * K_ + row];
  }
#endif
  __syncthreads();

  if (tid < 16) {
    float s = 0.f;
#pragma unroll
    for (int d = 0; d < D_; ++d) {
      float v = sA[d * 16 + tid];
      s += v * v;
    }
    sAnorm[tid] = s;
  }
  __syncthreads();

  int wave = tid >> 5;
  int lane = tid & 31;
  int hf = lane >> 4;    // K-half selector (lanes 16-31 hold K+2,K+3)
  int l16 = lane & 15;

  float bestV[8];
  int bestC[8];
#pragma unroll
  for (int r = 0; r < 8; ++r) { bestV[r] = 3.4e38f; bestC[r] = 0; }

  for (int tt = 0; tt < 4; ++tt) {
    int c0 = (wave * 4 + tt) * 16;
    int col = c0 + l16;
    const float* cbc = cb + (long)col * D_;
    v8f acc = {};
#pragma unroll
    for (int kk = 0; kk < D_; kk += 4) {
      int k2 = kk + 2 * hf;
      v2f a, bv;
      a.x = sA[(k2 + 0) * 16 + l16];
      a.y = sA[(k2 + 1) * 16 + l16];
      bv.x = cbc[k2 + 0];
      bv.y = cbc[k2 + 1];
      acc = wmma_f32(a, bv, acc);
    }
    union { v8f v; float f[8]; } u;
    u.v = acc;
#pragma unroll
    for (int r = 0; r < 8; ++r) {
      float d2 = sAnorm[r + 8 * hf] - 2.f * u.f[r] + cnorm[col];
      int c = col;
#pragma unroll
      for (int m = 8; m >= 1; m >>= 1) {
        float ov = __shfl_xor(d2, m, 32);
        int oc = __shfl_xor(c, m, 32);
        if (ov < d2 || (ov == d2 && oc < c)) { d2 = ov; c = oc; }
      }
      if (d2 < bestV[r] || (d2 == bestV[r] && c < bestC[r])) {
        bestV[r] = d2;
        bestC[r] = c;
      }
    }
  }
  if (l16 == 0) {
#pragma unroll
    for (int r = 0; r < 8; ++r) {
      sVal[wave][r + 8 * hf] = bestV[r];
      sCol[wave][r + 8 * hf] = bestC[r];
    }
  }
  __syncthreads();
  if (tid < 16) {
    float bv = sVal[0][tid];
    int bc = sCol[0][tid];
    for (int w2 = 1; w2 < 8; ++w2) {
      float v = sVal[w2][tid];
      int c = sCol[w2][tid];
      if (v < bv || (v == bv && c < bc)) { bv = v; bc = c; }
    }
    codes[n0 + tid] = bc;
  }
}

// ---------------- gather zq + perturbed zq ----------------
__global__ void zq_gather_k(const float* __restrict__ cb, const int* __restrict__ codes,
                            const float* __restrict__ noise, float* __restrict__ zq,
                            float* __restrict__ zqp) {
  int idx = blockIdx.x * blockDim.x + threadIdx.x;
  if (idx >= B_ * D_ * K_) return;
  int t = idx % K_;
  int d = (idx / K_) % D_;
  int b = idx / (K_ * D_);
  float v = cb[(long)codes[b * K_ + t] * D_ + d];
  zq[idx] = v;
  zqp[idx] = v + EPS_LIP_ * noise[idx];
}

// ---------------- sum of squared differences -> atomic accumulator ----------------
__global__ void mse_acc_k(const float* __restrict__ a, const float* __restrict__ b,
                          int n, float* __restrict__ acc) {
  __shared__ float sbuf[256];
  int idx = blockIdx.x * 256 + threadIdx.x;
  float d = 0.f;
  if (idx < n) {
    float t = a[idx] - b[idx];
    d = t * t;
  }
  float s = blockReduceSum(d, sbuf);
  if (threadIdx.x == 0) atomicAdd(acc, s);
}

// ---------------- log-volume branch ----------------
__global__ void lv_k(const float* __restrict__ x, const float* __restrict__ hd2,
                     const float* __restrict__ Wlv, const float* __restrict__ blv,
                     float* __restrict__ lv, float* __restrict__ tgt) {
  __shared__ float sbuf[256];
  int b = blockIdx.x;
  int tid = threadIdx.x;
  float sx = 0.f;
  for (int t = tid; t < L_; t += 256) sx += x[(long)b * CIN_ * L_ + t];  // channel 0
  float sxr = blockReduceSum(sx, sbuf);
  __syncthreads();
  float sl = 0.f;
  for (int i = tid; i < D_ * L_; i += 256) {
    int c = i / L_;
    sl += hd2[(long)b * D_ * L_ + i] * Wlv[c];
  }
  float slr = blockReduceSum(sl, sbuf);
  if (tid == 0) {
    tgt[b] = logf(sxr + 1e-7f);
    lv[b] = slr / (float)L_ + blv[0];
  }
}

__global__ void lvloss_k(const float* __restrict__ lv, const float* __restrict__ tgt,
                         float* __restrict__ out) {
  int idx = blockIdx.x * 256 + threadIdx.x;
  if (idx >= LV_SZ) return;
  int i = idx / B_, j = idx % B_;
  float d = tgt[i] - lv[j];
  out[Y_SZ + idx] = d * d;
}

// ---------------- softmax over length dim ----------------
__global__ void softmax_k(const float* __restrict__ yraw, float* __restrict__ y) {
  __shared__ float sbuf[256];
  int row = blockIdx.x;  // B*C rows of L
  const float* r = yraw + (long)row * L_;
  float* o = y + (long)row * L_;
  int tid = threadIdx.x;
  float m = -3.4e38f;
  for (int t = tid; t < L_; t += 256) m = fmaxf(m, r[t]);
  sbuf[tid] = m;
  __syncthreads();
  for (int s = 128; s > 0; s >>= 1) {
    if (tid < s) sbuf[tid] = fmaxf(sbuf[tid], sbuf[tid + s]);
    __syncthreads();
  }
  m = sbuf[0];
  __syncthreads();
  float sum = 0.f;
  for (int t = tid; t < L_; t += 256) sum += __expf(r[t] - m);
  float tot = blockReduceSum(sum, sbuf);
  float inv = 1.0f / tot;
  for (int t = tid; t < L_; t += 256) o[t] = __expf(r[t] - m) * inv;
}

// ---------------- pairwise decode distance via WMMA Gram matrix ----------------
// dist[i,j] = (||y_i||^2 + ||y_j||^2 - 2 * y_i.y_j) / (C*L)
// grid: 16 blocks (4x4 tiles of the 64x64 output), 1 wave per block.
__global__ void gram_dist_k(const float* __restrict__ Y, const float* __restrict__ snorm,
                            float* __restrict__ dist) {
  int ti = blockIdx.x & 3;
  int tj = blockIdx.x >> 2;
  int i0 = ti * 16, j0 = tj * 16;
  int lane = threadIdx.x & 31;
  int hf = lane >> 4;
  int l16 = lane & 15;
  const float* ra = Y + (long)(i0 + l16) * CL_;
  const float* rb = Y + (long)(j0 + l16) * CL_;
  v8f acc = {};
  for (int k = 0; k < CL_; k += 4) {
    int k2 = k + 2 * hf;
    if ((k & 63) == 0) {
      __builtin_prefetch(&ra[k2 + 256], 0, 1);
      __builtin_prefetch(&rb[k2 + 256], 0, 1);
    }
    v2f a, bv;
    a.x = ra[k2];
    a.y = ra[k2 + 1];
    bv.x = rb[k2];
    bv.y = rb[k2 + 1];
    acc = wmma_f32(a, bv, acc);
  }
  union { v8f v; float f[8]; } u;
  u.v = acc;
  const float invCL = 1.0f / (float)CL_;
#pragma unroll
  for (int r = 0; r < 8; ++r) {
    int i = i0 + r + 8 * hf;
    int j = j0 + l16;
    dist[i * B_ + j] = (snorm[i] + snorm[j] - 2.f * u.f[r]) * invCL;
  }
}

// ---------------- code reconstruction variance loss ----------------
__global__ void crv_k(const int* __restrict__ codes, const float* __restrict__ dist,
                      float* __restrict__ acc) {
  __shared__ float sn[256];
  __shared__ float sd[256];
  int k = blockIdx.x;
  int tid = threadIdx.x;
  float num = 0.f, den = 0.f;
  for (int p = tid; p < B_ * B_; p += 256) {
    int i = p / B_, j = p % B_;
    if (codes[i * K_ + k] == codes[j * K_ + k]) {
      num += dist[p];
      den += 1.f;
    }
  }
  sn[tid] = num;
  sd[tid] = den;
  __syncthreads();
  for (int s = 128; s > 0; s >>= 1) {
    if (tid < s) {
      sn[tid] += sn[tid + s];
      sd[tid] += sd[tid + s];
    }
    __syncthreads();
  }
  if (tid == 0) atomicAdd(acc, (sn[0] / (sd[0] + 1e-6f)) * (1.0f / (float)K_));
}

// ---------------- init / finalize ----------------
__global__ void init_k(float* __restrict__ accs) {
  if (threadIdx.x < 4) accs[threadIdx.x] = 0.f;
}

__global__ void finalize_k(const float* __restrict__ accs, float* __restrict__ out) {
  // accs[0]=sum(zq-z)^2, accs[1]=sum(yraw-y2)^2, accs[2]=crv
  out[Y_SZ + LV_SZ + 0] = BETA_ * accs[0] / (float)(B_ * D_ * K_);
  out[Y_SZ + LV_SZ + 1] =
      accs[1] / ((float)Y_SZ * (EPS_LIP_ * EPS_LIP_)) + accs[2];
}

// ---------------- host launcher ----------------
extern "C" void kernel_launch(void* const* d_in, const int* in_sizes, int n_in,
                              void* d_out, int out_size, void* d_ws, size_t ws_size,
                              hipStream_t stream) {
  const float* x   = (const float*)d_in[0];
  const float* We1 = (const float*)d_in[1];
  const float* be1 = (const float*)d_in[2];
  const float* We2 = (const float*)d_in[3];
  const float* be2 = (const float*)d_in[4];
  const float* We3 = (const float*)d_in[5];
  const float* be3 = (const float*)d_in[6];
  const float* Wd1 = (const float*)d_in[7];
  const float* bd1 = (const float*)d_in[8];
  const float* Wd2 = (const float*)d_in[9];
  const float* bd2 = (const float*)d_in[10];
  const float* Wd3 = (const float*)d_in[11];
  const float* bd3 = (const float*)d_in[12];
  const float* cb  = (const float*)d_in[13];
  const float* Wlv = (const float*)d_in[14];
  const float* blv = (const float*)d_in[15];
  const float* noise = (const float*)d_in[16];
  float* out = (float*)d_out;
  float* ws = (float*)d_ws;

  // workspace layout (floats)
  float* h1    = ws + 0;          // 2,097,152  [B,64,512]
  float* h2e   = ws + 2097152;    // 2,097,152  [B,128,256]
  float* z     = ws + 4194304;    // 1,048,576  [B,64,256]
  float* zq    = ws + 5242880;    // 1,048,576
  float* zqp   = ws + 6291456;    // 1,048,576
  float* hd1   = ws + 7340032;    // 4,194,304  [B,128,512]
  float* hd2   = ws + 11534336;   // 4,194,304  [B,64,1024]
  float* yraw  = ws + 15728640;   // 524,288    [B,8,1024]
  float* y2raw = ws + 16252928;   // 524,288
  float* cnormp = ws + 16777216;  // 512
  int*   codes  = (int*)(ws + 16777728);  // 16,384 ints
  float* snorm  = ws + 16794112;  // 64
  float* dist   = ws + 16794176;  // 4,096
  float* lv     = ws + 16798272;  // 64
  float* tgt    = ws + 16798336;  // 64
  float* accs   = ws + 16798400;  // 4

  dim3 blk(256);
  init_k<<<1, 64, 0, stream>>>(accs);

  // encoder
  conv1d_t<CIN_, 4, 2, 1, 1><<<(B_ * 64 * 512 + 255) / 256, blk, 0, stream>>>(
      x, We1, be1, h1, 1024, 64, 512);
  conv1d_t<64, 4, 2, 1, 1><<<(B_ * 128 * 256 + 255) / 256, blk, 0, stream>>>(
      h1, We2, be2, h2e, 512, 128, 256);
  conv1d_t<128, 3, 1, 1, 0><<<(B_ * 64 * 256 + 255) / 256, blk, 0, stream>>>(
      h2e, We3, be3, z, 256, D_, 256);

  // VQ (TDM-staged A tile + WMMA GEMM distances + argmin)
  rownorm_k<<<(E_ + 255) / 256, blk, 0, stream>>>(cb, cnormp, E_, D_);
  vq_assign_k<<<(B_ * K_) / 16, blk, 0, stream>>>(z, cb, cnormp, codes);
  zq_gather_k<<<(B_ * D_ * K_ + 255) / 256, blk, 0, stream>>>(cb, codes, noise, zq, zqp);
  mse_acc_k<<<(B_ * D_ * K_ + 255) / 256, blk, 0, stream>>>(zq, z, B_ * D_ * K_, accs + 0);

  // decoder pass 1 (zq)
  convT1d_t<D_, 4, 2, 1, 1><<<(B_ * 128 * 512 + 255) / 256, blk, 0, stream>>>(
      zq, Wd1, bd1, hd1, 256, 128, 512);
  convT1d_t<128, 4, 2, 1, 1><<<(B_ * 64 * 1024 + 255) / 256, blk, 0, stream>>>(
      hd1, Wd2, bd2, hd2, 512, 64, 1024);
  conv1d_t<64, 3, 1, 1, 0><<<(B_ * 8 * 1024 + 255) / 256, blk, 0, stream>>>(
      hd2, Wd3, bd3, yraw, 1024, CIN_, 1024);

  // log-volume branch (needs pass-1 hd2 before it is reused)
  lv_k<<<B_, blk, 0, stream>>>(x, hd2, Wlv, blv, lv, tgt);
  lvloss_k<<<(LV_SZ + 255) / 256, blk, 0, stream>>>(lv, tgt, out);

  // decoder pass 2 (perturbed), reuse hd1/hd2
  convT1d_t<D_, 4, 2, 1, 1><<<(B_ * 128 * 512 + 255) / 256, blk, 0, stream>>>(
      zqp, Wd1, bd1, hd1, 256, 128, 512);
  convT1d_t<128, 4, 2, 1, 1><<<(B_ * 64 * 1024 + 255) / 256, blk, 0, stream>>>(
      hd1, Wd2, bd2, hd2, 512, 64, 1024);
  conv1d_t<64, 3, 1, 1, 0><<<(B_ * 8 * 1024 + 255) / 256, blk, 0, stream>>>(
      hd2, Wd3, bd3, y2raw, 1024, CIN_, 1024);
  mse_acc_k<<<(Y_SZ + 255) / 256, blk, 0, stream>>>(yraw, y2raw, Y_SZ, accs + 1);

  // softmax -> y (lives at start of d_out)
  softmax_k<<<B_ * CIN_, blk, 0, stream>>>(yraw, out);

  // pairwise distance via WMMA Gram matrix
  rownorm_block_k<<<B_, blk, 0, stream>>>(out, snorm, CL_);
  gram_dist_k<<<16, dim3(32), 0, stream>>>(out, snorm, dist);

  // code reconstruction variance
  crv_k<<<K_, blk, 0, stream>>>(codes, dist, accs + 2);

  finalize_k<<<1, 1, 0, stream>>>(accs, out);
}